// QuantumNet_35235911696436
// MI455X (gfx1250) — compile-verified
//
#include <hip/hip_runtime.h>
#include <math.h>

// ---------------------------------------------------------------------------
// QuantumNet fused kernel for gfx1250 (MI455X).
//   pre:   [B,512] x [4,512]^T  via V_WMMA_F32_16X16X4_F32 (f32 matrix pipe)
//   mid:   per-row 4-qubit statevector circuit (16 complex amps in VGPRs)
//   post:  [B,4] x [2,4]^T + bias -> [B,2]
// Memory-bound on the 64MB activation stream; one pass, no global scratch.
// ---------------------------------------------------------------------------

typedef __attribute__((ext_vector_type(2))) float v2f;
typedef __attribute__((ext_vector_type(8))) float v8f;

#define KDIM 512
#define DIM 16

// Apply a complex 2x2 gate on qubit Q (Q=0 is MSB -> state bit (3-Q)).
template<int Q>
__device__ __forceinline__ void gate1(float* sr, float* si,
    float g00r, float g00i, float g01r, float g01i,
    float g10r, float g10i, float g11r, float g11i)
{
    constexpr int m = 1 << (3 - Q);
#pragma unroll
    for (int i = 0; i < DIM; ++i) {
        if (i & m) continue;
        const int j = i | m;
        const float ar = sr[i], ai = si[i], br = sr[j], bi = si[j];
        sr[i] = g00r * ar - g00i * ai + g01r * br - g01i * bi;
        si[i] = g00r * ai + g00i * ar + g01r * bi + g01i * br;
        sr[j] = g10r * ar - g10i * ai + g11r * br - g11i * bi;
        si[j] = g10r * ai + g10i * ar + g11r * bi + g11i * br;
    }
}

template<int CQ, int TQ>
__device__ __forceinline__ void cnot(float* sr, float* si)
{
    constexpr int cm = 1 << (3 - CQ);
    constexpr int tm = 1 << (3 - TQ);
#pragma unroll
    for (int i = 0; i < DIM; ++i) {
        if ((i & cm) && !(i & tm)) {
            const int j = i | tm;
            const float tr = sr[i], ti = si[i];
            sr[i] = sr[j]; si[i] = si[j];
            sr[j] = tr;    si[j] = ti;
        }
    }
}

template<int Q>
__device__ __forceinline__ void applyH(float* sr, float* si)
{
    const float r = 0.70710678118654752440f;
    gate1<Q>(sr, si, r, 0.f, r, 0.f, r, 0.f, -r, 0.f);
}

template<int Q>
__device__ __forceinline__ void applyRY(float* sr, float* si, float th)
{
    const float c = cosf(0.5f * th);
    const float s = sinf(0.5f * th);
    gate1<Q>(sr, si, c, 0.f, -s, 0.f, s, 0.f, c, 0.f);
}

template<int Q>
__device__ __forceinline__ void applyRZ(float* sr, float* si, float th)
{
    const float c = cosf(0.5f * th);
    const float s = sinf(0.5f * th);
    // diag(e^{-i th/2}, e^{+i th/2})
    gate1<Q>(sr, si, c, -s, 0.f, 0.f, 0.f, 0.f, c, s);
}

template<int Q>
__device__ __forceinline__ void applyU3(float* sr, float* si, const float* u3p)
{
    const float th = u3p[Q * 3 + 0];
    const float ph = u3p[Q * 3 + 1];
    const float la = u3p[Q * 3 + 2];
    const float ct = cosf(0.5f * th), st = sinf(0.5f * th);
    const float elr = cosf(la),      eli = sinf(la);        // e^{i la}
    const float epr = cosf(ph),      epi = sinf(ph);        // e^{i ph}
    const float eqr = cosf(ph + la), eqi = sinf(ph + la);   // e^{i (ph+la)}
    gate1<Q>(sr, si,
             ct, 0.f,
             -elr * st, -eli * st,
             epr * st,  epi * st,
             eqr * ct,  eqi * ct);
}

__global__ __launch_bounds__(256) void qnet_fused_kernel(
    const float* __restrict__ x,       // [B,512]
    const float* __restrict__ pre_w,   // [4,512]
    const float* __restrict__ pre_b,   // [4]
    const float* __restrict__ u3p,     // [4,3]
    const float* __restrict__ post_w,  // [2,4]
    const float* __restrict__ post_b,  // [2]
    float* __restrict__ out,           // [B,2]
    int B)
{
    __shared__ float lds_pre[256 * 4];

    const int tid  = threadIdx.x;
    const int wave = tid >> 5;
    const int lane = tid & 31;
    const int blockRow0 = blockIdx.x * 256;

    const int laneHalf = lane >> 4;   // 0: K={0,1}  1: K={2,3}
    const int laneMod  = lane & 15;   // A: row m ; B: col n
    const int koff     = laneHalf * 2;

    // B fragment: column n>=4 contributes zero (branch-free, EXEC stays full)
    const int   nc     = laneMod & 3;
    const float bscale = (laneMod < 4) ? 1.0f : 0.0f;
    const float* brow  = pre_w + (size_t)nc * KDIM + koff;

    // ---- WMMA phase: two 16-row tiles per wave -> 256 rows per block ----
#pragma unroll
    for (int t = 0; t < 2; ++t) {
        const int row0 = blockRow0 + wave * 32 + t * 16;
        int arowIdx = row0 + laneMod;
        if (arowIdx > B - 1) arowIdx = B - 1;   // clamp keeps EXEC uniform
        const float* arow = x + (size_t)arowIdx * KDIM + koff;

        v8f acc = {};
#pragma unroll 8
        for (int k0 = 0; k0 < KDIM; k0 += 4) {
            v2f a  = *(const v2f*)(arow + k0);
            v2f bv = *(const v2f*)(brow + k0);
            bv *= bscale;
            // D = A(16x4,f32) * B(4x16,f32) + C
            acc = __builtin_amdgcn_wmma_f32_16x16x4_f32(
                false, a, false, bv, (short)0, acc, false, false);
        }

        // Valid columns n=0..3 live in lanes 0-3 (rows r) / 16-19 (rows r+8)
        if (laneMod < 4) {
#pragma unroll
            for (int r = 0; r < 8; ++r) {
                const int m = r + laneHalf * 8;
                lds_pre[(wave * 32 + t * 16 + m) * 4 + laneMod] = acc[r];
            }
        }
    }
    __syncthreads();

    // ---- Circuit phase: one batch row per thread ----
    const int grow = blockRow0 + tid;

    float ang[4];
#pragma unroll
    for (int q = 0; q < 4; ++q) {
        const float pre  = lds_pre[tid * 4 + q] + pre_b[q];
        const float q_in = tanhf(pre * 0.1f) * 1.57079632679489661923f;
        ang[q] = q_in;
    }
    const float ry0 = atanf(ang[0]), ry1 = atanf(ang[1]);
    const float ry2 = atanf(ang[2]), ry3 = atanf(ang[3]);
    const float rz0 = atanf(ang[0] * ang[0]), rz1 = atanf(ang[1] * ang[1]);
    const float rz2 = atanf(ang[2] * ang[2]), rz3 = atanf(ang[3] * ang[3]);

    float sr[DIM], si[DIM];
#pragma unroll
    for (int i = 0; i < DIM; ++i) { sr[i] = 0.f; si[i] = 0.f; }
    sr[0] = 1.f;

    applyH<0>(sr, si); applyH<1>(sr, si); applyH<2>(sr, si); applyH<3>(sr, si);

    applyRY<0>(sr, si, ry0); applyRY<1>(sr, si, ry1);
    applyRY<2>(sr, si, ry2); applyRY<3>(sr, si, ry3);

    applyRZ<0>(sr, si, rz0); applyRZ<1>(sr, si, rz1);
    applyRZ<2>(sr, si, rz2); applyRZ<3>(sr, si, rz3);

    // cnot ring, step 1
    cnot<0, 1>(sr, si); cnot<1, 2>(sr, si); cnot<2, 3>(sr, si); cnot<3, 0>(sr, si);
    // cnot ring, step 2
    cnot<0, 2>(sr, si); cnot<1, 3>(sr, si); cnot<2, 0>(sr, si); cnot<3, 1>(sr, si);

    applyU3<0>(sr, si, u3p); applyU3<1>(sr, si, u3p);
    applyU3<2>(sr, si, u3p); applyU3<3>(sr, si, u3p);

    // Z expectations per qubit
    float p[DIM];
#pragma unroll
    for (int i = 0; i < DIM; ++i) p[i] = sr[i] * sr[i] + si[i] * si[i];

    float e[4];
#pragma unroll
    for (int q = 0; q < 4; ++q) {
        const int m = 1 << (3 - q);
        float acc = 0.f;
#pragma unroll
        for (int i = 0; i < DIM; ++i)
            acc += (i & m) ? -p[i] : p[i];
        e[q] = acc;
    }

    // post projection: [2,4] @ e + b
    float o0 = post_b[0];
    float o1 = post_b[1];
#pragma unroll
    for (int q = 0; q < 4; ++q) {
        o0 += e[q] * post_w[q];
        o1 += e[q] * post_w[4 + q];
    }

    if (grow < B) {
        v2f ov; ov.x = o0; ov.y = o1;
        *(v2f*)(out + (size_t)grow * 2) = ov;
    }
}

extern "C" void kernel_launch(void* const* d_in, const int* in_sizes, int n_in,
                              void* d_out, int out_size, void* d_ws, size_t ws_size,
                              hipStream_t stream) {
    const float* x      = (const float*)d_in[0];  // [B,512]
    const float* pre_w  = (const float*)d_in[1];  // [4,512]
    const float* pre_b  = (const float*)d_in[2];  // [4]
    const float* u3p    = (const float*)d_in[3];  // [4,3]
    const float* post_w = (const float*)d_in[4];  // [2,4]
    const float* post_b = (const float*)d_in[5];  // [2]
    float* out          = (float*)d_out;          // [B,2]

    const int B = in_sizes[0] / KDIM;             // 32768
    const int blocks = (B + 255) / 256;           // 128

    qnet_fused_kernel<<<blocks, 256, 0, stream>>>(
        x, pre_w, pre_b, u3p, post_w, post_b, out, B);
}